// SVD_85194971283660
// MI455X (gfx1250) — compile-verified
//
#include <hip/hip_runtime.h>
#include <cstdint>

#define TPB 256

// ---------------------------------------------------------------------------
// CDNA5 async global<->LDS helpers (gfx1250, ASYNCcnt path)
// ---------------------------------------------------------------------------
template <int OFF>
__device__ __forceinline__ void asyncLoadB32(uint32_t ldsAddr, uint32_t gOff, uint64_t base) {
  // one dword per lane, global (saddr+vaddr) -> LDS (vdst)  (GVS mode; OFF applies to both sides)
  asm volatile("global_load_async_to_lds_b32 %0, %1, %2 offset:%3"
               :: "v"(ldsAddr), "v"(gOff), "s"(base), "n"(OFF)
               : "memory");
}

template <int OFF>
__device__ __forceinline__ void asyncStoreB32(uint32_t gOff, uint32_t ldsAddr, uint64_t base) {
  // one dword per lane, LDS (vsrc) -> global (saddr+vaddr); OFF on both sides
  asm volatile("global_store_async_from_lds_b32 %0, %1, %2 offset:%3"
               :: "v"(gOff), "v"(ldsAddr), "s"(base), "n"(OFF)
               : "memory");
}

#define WAIT_ASYNCCNT(n) asm volatile("s_wait_asynccnt %0" :: "n"(n) : "memory")
#define WAIT_DSCNT0()    asm volatile("s_wait_dscnt 0x0" ::: "memory")

// ---------------------------------------------------------------------------
// Branch-free 3x3 SVD (McAdams et al. style), det(U)=det(V)=+1, s0>=s1>=|s2|
// ---------------------------------------------------------------------------
__device__ __forceinline__ float nrRsqrt(float x) {
  float w = rsqrtf(x);
  return w * (1.5f - 0.5f * x * w * w);  // one Newton step, ~1e-7 rel err
}

__device__ __forceinline__ void condSwap(bool c, float& x, float& y) {
  float t = x; x = c ? y : x; y = c ? t : y;
}
__device__ __forceinline__ void condNegSwap(bool c, float& x, float& y) {
  float t = -x; x = c ? y : x; y = c ? t : y;
}

template <int X, int Y, int Z>
__device__ __forceinline__ void jacobiConj(float& s11, float& s21, float& s22,
                                           float& s31, float& s32, float& s33,
                                           float q[4]) {
  const float kGamma = 5.82842712474619f;   // 3 + 2*sqrt(2)
  const float kCstar = 0.923879532511287f;  // cos(pi/8)
  const float kSstar = 0.382683432365090f;  // sin(pi/8)

  float ch = 2.0f * (s11 - s22);
  float sh = s21;
  bool usew = (kGamma * sh * sh) < (ch * ch);
  float w = nrRsqrt(ch * ch + sh * sh);
  ch = usew ? w * ch : kCstar;
  sh = usew ? w * sh : kSstar;

  float a = ch * ch - sh * sh;
  float b = 2.0f * sh * ch;

  float t11 = s11, t21 = s21, t22 = s22, t31 = s31, t32 = s32, t33 = s33;
  s11 =  a * ( a * t11 + b * t21) + b * ( a * t21 + b * t22);
  s21 =  a * (-b * t11 + a * t21) + b * (-b * t21 + a * t22);
  s22 = -b * (-b * t11 + a * t21) + a * (-b * t21 + a * t22);
  s31 =  a * t31 + b * t32;
  s32 = -b * t31 + a * t32;
  s33 =  t33;

  float tmp[3];
  tmp[0] = q[0] * sh; tmp[1] = q[1] * sh; tmp[2] = q[2] * sh;
  sh *= q[3];
  q[0] *= ch; q[1] *= ch; q[2] *= ch; q[3] *= ch;
  q[Z] += sh;
  q[3] -= tmp[Z];
  q[X] += tmp[Y];
  q[Y] -= tmp[X];

  // cyclic permutation so the next rotation hits the next (p,q) pair
  float n11 = s22, n21 = s32, n22 = s33, n31 = s21, n32 = s31, n33 = s11;
  s11 = n11; s21 = n21; s22 = n22; s31 = n31; s32 = n32; s33 = n33;
}

__device__ __forceinline__ void qrGivens(float a1, float a2, float& ch, float& sh) {
  const float kEps = 1.0e-6f;
  float rho = sqrtf(a1 * a1 + a2 * a2);
  sh = (rho > kEps) ? a2 : 0.0f;
  ch = fabsf(a1) + fmaxf(rho, kEps);
  condSwap(a1 < 0.0f, sh, ch);
  float w = nrRsqrt(ch * ch + sh * sh);
  ch *= w; sh *= w;
}

__device__ __forceinline__ void svd3x3(const float f[9], float u[9], float sg[3], float vh[9]) {
  float a11 = f[0], a12 = f[1], a13 = f[2];
  float a21 = f[3], a22 = f[4], a23 = f[5];
  float a31 = f[6], a32 = f[7], a33 = f[8];

  // S = A^T A (lower triangle)
  float s11 = a11*a11 + a21*a21 + a31*a31;
  float s21 = a11*a12 + a21*a22 + a31*a32;
  float s22 = a12*a12 + a22*a22 + a32*a32;
  float s31 = a11*a13 + a21*a23 + a31*a33;
  float s32 = a12*a13 + a22*a23 + a32*a33;
  float s33 = a13*a13 + a23*a23 + a33*a33;

  float q[4] = {0.0f, 0.0f, 0.0f, 1.0f};
#pragma unroll
  for (int i = 0; i < 4; ++i) {
    jacobiConj<0, 1, 2>(s11, s21, s22, s31, s32, s33, q);
    jacobiConj<1, 2, 0>(s11, s21, s22, s31, s32, s33, q);
    jacobiConj<2, 0, 1>(s11, s21, s22, s31, s32, s33, q);
  }

  float qq = q[0]*q[0] + q[1]*q[1] + q[2]*q[2] + q[3]*q[3];
  float iq = nrRsqrt(qq);
  float qx = q[0]*iq, qy = q[1]*iq, qz = q[2]*iq, qw = q[3]*iq;

  float v11 = 1.0f - 2.0f*(qy*qy + qz*qz);
  float v12 = 2.0f*(qx*qy - qw*qz);
  float v13 = 2.0f*(qx*qz + qw*qy);
  float v21 = 2.0f*(qx*qy + qw*qz);
  float v22 = 1.0f - 2.0f*(qx*qx + qz*qz);
  float v23 = 2.0f*(qy*qz - qw*qx);
  float v31 = 2.0f*(qx*qz - qw*qy);
  float v32 = 2.0f*(qy*qz + qw*qx);
  float v33 = 1.0f - 2.0f*(qx*qx + qy*qy);

  // B = A * V
  float b11 = a11*v11 + a12*v21 + a13*v31;
  float b12 = a11*v12 + a12*v22 + a13*v32;
  float b13 = a11*v13 + a12*v23 + a13*v33;
  float b21 = a21*v11 + a22*v21 + a23*v31;
  float b22 = a21*v12 + a22*v22 + a23*v32;
  float b23 = a21*v13 + a22*v23 + a23*v33;
  float b31 = a31*v11 + a32*v21 + a33*v31;
  float b32 = a31*v12 + a32*v22 + a33*v32;
  float b33 = a31*v13 + a32*v23 + a33*v33;

  // sort columns descending by norm; negated swaps keep det = +1
  float r1 = b11*b11 + b21*b21 + b31*b31;
  float r2 = b12*b12 + b22*b22 + b32*b32;
  float r3 = b13*b13 + b23*b23 + b33*b33;
  bool c = r1 < r2;
  condNegSwap(c, b11, b12); condNegSwap(c, v11, v12);
  condNegSwap(c, b21, b22); condNegSwap(c, v21, v22);
  condNegSwap(c, b31, b32); condNegSwap(c, v31, v32);
  condSwap(c, r1, r2);
  c = r1 < r3;
  condNegSwap(c, b11, b13); condNegSwap(c, v11, v13);
  condNegSwap(c, b21, b23); condNegSwap(c, v21, v23);
  condNegSwap(c, b31, b33); condNegSwap(c, v31, v33);
  condSwap(c, r1, r3);
  c = r2 < r3;
  condNegSwap(c, b12, b13); condNegSwap(c, v12, v13);
  condNegSwap(c, b22, b23); condNegSwap(c, v22, v23);
  condNegSwap(c, b32, b33); condNegSwap(c, v32, v33);

  // Givens-QR of B: B = Q R, U = Q, sigma = diag(R)
  float ch1, sh1, ch2, sh2, ch3, sh3, aa, bb;
  qrGivens(b11, b21, ch1, sh1);
  aa = 1.0f - 2.0f*sh1*sh1; bb = 2.0f*ch1*sh1;
  float r11 =  aa*b11 + bb*b21, r12 =  aa*b12 + bb*b22, r13 =  aa*b13 + bb*b23;
  float r21 = -bb*b11 + aa*b21, r22 = -bb*b12 + aa*b22, r23 = -bb*b13 + aa*b23;
  float r31 = b31, r32 = b32, r33 = b33;

  qrGivens(r11, r31, ch2, sh2);
  aa = 1.0f - 2.0f*sh2*sh2; bb = 2.0f*ch2*sh2;
  b11 =  aa*r11 + bb*r31; b12 =  aa*r12 + bb*r32; b13 =  aa*r13 + bb*r33;
  b21 = r21; b22 = r22; b23 = r23;
  b31 = -bb*r11 + aa*r31; b32 = -bb*r12 + aa*r32; b33 = -bb*r13 + aa*r33;

  qrGivens(b22, b32, ch3, sh3);
  aa = 1.0f - 2.0f*sh3*sh3; bb = 2.0f*ch3*sh3;
  r11 = b11; r12 = b12; r13 = b13;
  r21 =  aa*b21 + bb*b31; r22 =  aa*b22 + bb*b32; r23 =  aa*b23 + bb*b33;
  r31 = -bb*b21 + aa*b31; r32 = -bb*b22 + aa*b32; r33 = -bb*b23 + aa*b33;

  float sh12 = sh1*sh1, sh22 = sh2*sh2, sh32 = sh3*sh3;
  u[0] = (-1.0f + 2.0f*sh12) * (-1.0f + 2.0f*sh22);
  u[1] = 4.0f*ch2*ch3*(-1.0f + 2.0f*sh12)*sh2*sh3 + 2.0f*ch1*sh1*(-1.0f + 2.0f*sh32);
  u[2] = 4.0f*ch1*ch3*sh1*sh3 - 2.0f*ch2*(-1.0f + 2.0f*sh12)*sh2*(-1.0f + 2.0f*sh32);
  u[3] = 2.0f*ch1*sh1*(1.0f - 2.0f*sh22);
  u[4] = -8.0f*ch1*ch2*ch3*sh1*sh2*sh3 + (-1.0f + 2.0f*sh12)*(-1.0f + 2.0f*sh32);
  u[5] = -2.0f*ch3*sh3 + 4.0f*sh1*(ch3*sh1*sh3 + ch1*ch2*sh2*(-1.0f + 2.0f*sh32));
  u[6] = 2.0f*ch2*sh2;
  u[7] = 2.0f*ch3*(1.0f - 2.0f*sh22)*sh3;
  u[8] = (-1.0f + 2.0f*sh22) * (-1.0f + 2.0f*sh32);

  sg[0] = r11; sg[1] = r22; sg[2] = r33;

  // Vh = V^T (row i of Vh = column i of V)
  vh[0] = v11; vh[1] = v21; vh[2] = v31;
  vh[3] = v12; vh[4] = v22; vh[5] = v32;
  vh[6] = v13; vh[7] = v23; vh[8] = v33;
}

// ---------------------------------------------------------------------------
// Kernel: async-staged, barrier-free (all LDS regions are wave-private)
// ---------------------------------------------------------------------------
__global__ void __launch_bounds__(TPB)
svd3_batch_kernel(const float* __restrict__ F, float* __restrict__ out, int N) {
  __shared__ float ldsA[TPB * 9];  // F-in / U / Vh staging (per-wave 288-dword slices)
  __shared__ float ldsB[TPB * 3];  // sigma staging (per-wave 96-dword slices)

  const int tid  = threadIdx.x;
  const int lane = tid & 31;
  const int wv   = tid >> 5;
  const unsigned mBase = (unsigned)blockIdx.x * TPB;
  const unsigned N9  = 9u * (unsigned)N;
  const unsigned N12 = 12u * (unsigned)N;

  float f[9], u[9], sg[3], vh[9];
  const bool full = (mBase + TPB <= (unsigned)N);

  if (full) {
    const unsigned waveDw  = 288u * (unsigned)wv + (unsigned)lane;  // 9-wide slice
    const unsigned waveDw3 = 96u * (unsigned)wv + (unsigned)lane;   // 3-wide slice
    const uint32_t aLds = (uint32_t)(uintptr_t)ldsA + 4u * waveDw;
    const uint32_t bLds = (uint32_t)(uintptr_t)ldsB + 4u * waveDw3;
    const uint64_t fBase = (uint64_t)(uintptr_t)F;
    const uint64_t oBase = (uint64_t)(uintptr_t)out;

    // ---- async fill: 9 coalesced wave-rows, global -> LDS (ASYNCcnt) ----
    const uint32_t gIn = 4u * (9u * mBase + waveDw);
    asyncLoadB32<0>(aLds, gIn, fBase);    asyncLoadB32<128>(aLds, gIn, fBase);
    asyncLoadB32<256>(aLds, gIn, fBase);  asyncLoadB32<384>(aLds, gIn, fBase);
    asyncLoadB32<512>(aLds, gIn, fBase);  asyncLoadB32<640>(aLds, gIn, fBase);
    asyncLoadB32<768>(aLds, gIn, fBase);  asyncLoadB32<896>(aLds, gIn, fBase);
    asyncLoadB32<1024>(aLds, gIn, fBase);
    WAIT_ASYNCCNT(0);  // this wave's slice is resident

#pragma unroll
    for (int k = 0; k < 9; ++k) f[k] = ldsA[9 * tid + k];  // conflict-free (9 odd)

    svd3x3(f, u, sg, vh);

    // ---- U: stage into own slice, drain async LDS -> global ----
#pragma unroll
    for (int k = 0; k < 9; ++k) ldsA[9 * tid + k] = u[k];
    WAIT_DSCNT0();
    const uint32_t gU = 4u * (9u * mBase + waveDw);
    asyncStoreB32<0>(gU, aLds, oBase);    asyncStoreB32<128>(gU, aLds, oBase);
    asyncStoreB32<256>(gU, aLds, oBase);  asyncStoreB32<384>(gU, aLds, oBase);
    asyncStoreB32<512>(gU, aLds, oBase);  asyncStoreB32<640>(gU, aLds, oBase);
    asyncStoreB32<768>(gU, aLds, oBase);  asyncStoreB32<896>(gU, aLds, oBase);
    asyncStoreB32<1024>(gU, aLds, oBase);

    // ---- sigma: separate buffer so it overlaps the U drain ----
#pragma unroll
    for (int k = 0; k < 3; ++k) ldsB[3 * tid + k] = sg[k];
    WAIT_DSCNT0();
    const uint32_t gS = 4u * (N9 + 3u * mBase + waveDw3);
    asyncStoreB32<0>(gS, bLds, oBase);
    asyncStoreB32<128>(gS, bLds, oBase);
    asyncStoreB32<256>(gS, bLds, oBase);

    // async stores complete in order: cnt<=3 => the 9 U-stores drained ldsA
    WAIT_ASYNCCNT(3);

    // ---- Vh: reuse slice A ----
#pragma unroll
    for (int k = 0; k < 9; ++k) ldsA[9 * tid + k] = vh[k];
    WAIT_DSCNT0();
    const uint32_t gV = 4u * (N12 + 9u * mBase + waveDw);
    asyncStoreB32<0>(gV, aLds, oBase);    asyncStoreB32<128>(gV, aLds, oBase);
    asyncStoreB32<256>(gV, aLds, oBase);  asyncStoreB32<384>(gV, aLds, oBase);
    asyncStoreB32<512>(gV, aLds, oBase);  asyncStoreB32<640>(gV, aLds, oBase);
    asyncStoreB32<768>(gV, aLds, oBase);  asyncStoreB32<896>(gV, aLds, oBase);
    asyncStoreB32<1024>(gV, aLds, oBase);
    // S_ENDPGM performs an implicit wait-idle; no further LDS reuse.
  } else {
    // ragged tail block: guarded scalar path
    const unsigned m = mBase + (unsigned)tid;
    if (m < (unsigned)N) {
#pragma unroll
      for (int k = 0; k < 9; ++k) f[k] = F[9u * m + k];
    } else {
#pragma unroll
      for (int k = 0; k < 9; ++k) f[k] = (k % 4 == 0) ? 1.0f : 0.0f;  // identity
    }
    svd3x3(f, u, sg, vh);
    if (m < (unsigned)N) {
#pragma unroll
      for (int k = 0; k < 9; ++k) out[9u * m + k] = u[k];
#pragma unroll
      for (int k = 0; k < 3; ++k) out[N9 + 3u * m + k] = sg[k];
#pragma unroll
      for (int k = 0; k < 9; ++k) out[N12 + 9u * m + k] = vh[k];
    }
  }
}

extern "C" void kernel_launch(void* const* d_in, const int* in_sizes, int n_in,
                              void* d_out, int out_size, void* d_ws, size_t ws_size,
                              hipStream_t stream) {
  (void)n_in; (void)out_size; (void)d_ws; (void)ws_size;
  const float* F = (const float*)d_in[0];
  float* out = (float*)d_out;
  const int N = in_sizes[0] / 9;
  const int blocks = (N + TPB - 1) / TPB;
  svd3_batch_kernel<<<blocks, TPB, 0, stream>>>(F, out, N);
}